// SSIM_154618823017
// MI455X (gfx1250) — compile-verified
//
#include <hip/hip_runtime.h>

typedef float v2f __attribute__((ext_vector_type(2)));
typedef float v8f __attribute__((ext_vector_type(8)));

// ---- problem constants ----
#define IMG_H   512
#define IMG_W   512
#define PLANES  48            // 16 batch * 3 channels
#define TILE_H  32
#define TILE_W  64
#define NBLK    (PLANES * (IMG_H / TILE_H) * (IMG_W / TILE_W))   // 6144
#define NPIX    (PLANES * IMG_H * IMG_W)                         // 12582912

// ---- LDS tiling ----
#define RAW_ROWS   48         // 3 bands of 16: covers TILE_H+10 halo rows
#define RAW_COLS   74         // TILE_W + 10 halo cols (valid region)
#define RAW_STRIDE 82         // padded; even (b64 align), conflict-friendly
#define H_STRIDE   76         // padded stride for blurred planes

#define C1 1.0e-4f
#define C2 9.0e-4f

// normalized gaussian, sigma=1.5, 11 taps
#define G0 0.00102838f
#define G1 0.00759878f
#define G2 0.03600080f
#define G3 0.10936070f
#define G4 0.21300550f
#define G5 0.26601170f

__device__ __forceinline__ v8f wmma_f32(v2f a, v2f b, v8f c) {
    return __builtin_amdgcn_wmma_f32_16x16x4_f32(
        /*neg_a=*/false, a, /*neg_b=*/false, b,
        /*c_mod=*/(short)0, c, /*reuse_a=*/false, /*reuse_b=*/false);
}

__global__ __launch_bounds__(256)
void ssim_tile_kernel(const float* __restrict__ img1,
                      const float* __restrict__ img2,
                      float* __restrict__ ws) {
    __shared__ float raw1s[RAW_ROWS * RAW_STRIDE];
    __shared__ float raw2s[RAW_ROWS * RAW_STRIDE];
    __shared__ float Hs[5][RAW_ROWS * H_STRIDE];
    __shared__ float glds[16];
    __shared__ float wsum[8];

    const int tid = threadIdx.x;
    const int b   = blockIdx.x;
    const int plane = b >> 7;        // 128 tiles per 512x512 plane
    const int rem   = b & 127;
    const int rt    = rem >> 3;      // 16 row tiles of 32
    const int ct    = rem & 7;       // 8 col tiles of 64
    const int gr0   = rt * TILE_H;
    const int gc0   = ct * TILE_W;

    const float* __restrict__ p1 = img1 + (size_t)plane * (IMG_H * IMG_W);
    const float* __restrict__ p2 = img2 + (size_t)plane * (IMG_H * IMG_W);

    // gaussian tap table (indices 11..15 are zero pads for clamped lookups)
    if (tid == 0) {
        glds[0] = G0;  glds[1] = G1;  glds[2] = G2;  glds[3] = G3;
        glds[4] = G4;  glds[5] = G5;  glds[6] = G4;  glds[7] = G3;
        glds[8] = G2;  glds[9] = G1;  glds[10] = G0;
        glds[11] = 0.f; glds[12] = 0.f; glds[13] = 0.f; glds[14] = 0.f;
        glds[15] = 0.f;
    }

    // ---- stage raw tiles (with halo) into LDS; zero-fill pad + OOB (SAME) ----
    for (int i = tid; i < RAW_ROWS * RAW_STRIDE; i += 256) {
        const int r  = i / RAW_STRIDE;
        const int c  = i - r * RAW_STRIDE;
        const int gr = gr0 - 5 + r;
        const int gc = gc0 - 5 + c;
        const bool ok = ((unsigned)gr < (unsigned)IMG_H) &&
                        ((unsigned)gc < (unsigned)IMG_W) && (c < RAW_COLS);
        const int gi = (gr << 9) + gc;
        raw1s[i] = ok ? p1[gi] : 0.f;
        raw2s[i] = ok ? p2[gi] : 0.f;
    }
    __syncthreads();

    const int lane = tid & 31;
    const int half = lane >> 4;
    const int lr   = lane & 15;
    const int wv   = tid >> 5;

    // ---- per-lane constant Gaussian operand: W[j][i] = g[j-i] ----
    // Serves as B in the horizontal pass and as A in the vertical pass
    // (identical per-lane register image for both layouts).
    v2f cf[7];
#pragma unroll
    for (int k = 0; k < 7; ++k) {
        const int j0 = 4 * k + 2 * half;
        const int d0 = j0 - lr;
        const int d1 = d0 + 1;
        const int i0c = ((unsigned)d0 <= 10u) ? d0 : 11;
        const int i1c = ((unsigned)d1 <= 10u) ? d1 : 11;
        cf[k].x = glds[i0c];
        cf[k].y = glds[i1c];
    }

    const v8f vzero = {0.f, 0.f, 0.f, 0.f, 0.f, 0.f, 0.f, 0.f};

    // ---- horizontal pass: Out(16x16) = In(16x26) x Wh(26x16) via 7 K=4 WMMAs ----
    // 12 tiles (3 row bands x 4 col tiles) over the 48x64 H-region.
    for (int t = wv; t < 12; t += 8) {
        const int row0 = (t >> 2) * 16;
        const int col0 = (t & 3) * 16;
        v8f hacc[5];
#pragma unroll
        for (int q = 0; q < 5; ++q) hacc[q] = vzero;

#pragma unroll
        for (int k = 0; k < 7; ++k) {
            const int ai = (row0 + lr) * RAW_STRIDE + col0 + 4 * k + 2 * half;
            v2f a1; a1.x = raw1s[ai]; a1.y = raw1s[ai + 1];
            v2f a2; a2.x = raw2s[ai]; a2.y = raw2s[ai + 1];
            const v2f bb  = cf[k];
            const v2f a11 = a1 * a1;
            const v2f a22 = a2 * a2;
            const v2f a12 = a1 * a2;
            hacc[0] = wmma_f32(a1,  bb, hacc[0]);
            hacc[1] = wmma_f32(a2,  bb, hacc[1]);
            hacc[2] = wmma_f32(a11, bb, hacc[2]);
            hacc[3] = wmma_f32(a22, bb, hacc[3]);
            hacc[4] = wmma_f32(a12, bb, hacc[4]);
        }

        // D layout: VGPR vr -> row (vr + 8*half), col lr
#pragma unroll
        for (int q = 0; q < 5; ++q) {
#pragma unroll
            for (int vr = 0; vr < 8; ++vr) {
                Hs[q][(row0 + vr + 8 * half) * H_STRIDE + col0 + lr] = hacc[q][vr];
            }
        }
    }
    __syncthreads();

    // ---- vertical pass: Out(16x16) = Wv(16x26) x H(26x16); one tile per wave ----
    const int i0    = (wv >> 2) * 16;   // 2 row bands of 16 -> 32 out rows
    const int col0v = (wv & 3) * 16;    // 4 col tiles of 16 -> 64 out cols
    v8f vacc[5];
#pragma unroll
    for (int q = 0; q < 5; ++q) vacc[q] = vzero;

#pragma unroll
    for (int q = 0; q < 5; ++q) {
#pragma unroll
        for (int k = 0; k < 7; ++k) {
            const int bi = (i0 + 4 * k + 2 * half) * H_STRIDE + col0v + lr;
            v2f bb;
            bb.x = Hs[q][bi];
            bb.y = Hs[q][bi + H_STRIDE];
            vacc[q] = wmma_f32(cf[k], bb, vacc[q]);
        }
    }

    // ---- elementwise SSIM on aligned accumulators, then reduce ----
    float lsum = 0.f;
#pragma unroll
    for (int e = 0; e < 8; ++e) {
        const float mu1 = vacc[0][e];
        const float mu2 = vacc[1][e];
        const float mu1sq = mu1 * mu1;
        const float mu2sq = mu2 * mu2;
        const float mu12  = mu1 * mu2;
        const float s11 = vacc[2][e] - mu1sq;
        const float s22 = vacc[3][e] - mu2sq;
        const float s12 = vacc[4][e] - mu12;
        const float num = (2.f * mu12 + C1) * (2.f * s12 + C2);
        const float den = (mu1sq + mu2sq + C1) * (s11 + s22 + C2);
        lsum += num / den;
    }

    // fixed-order wave reduction (deterministic)
#pragma unroll
    for (int off = 16; off > 0; off >>= 1)
        lsum += __shfl_xor(lsum, off, 32);
    if (lane == 0) wsum[wv] = lsum;
    __syncthreads();
    if (tid == 0) {
        float s = 0.f;
#pragma unroll
        for (int i = 0; i < 8; ++i) s += wsum[i];
        ws[blockIdx.x] = s;
    }
}

__global__ __launch_bounds__(256)
void ssim_reduce_kernel(const float* __restrict__ ws, float* __restrict__ out) {
    __shared__ float sb[256];
    const int tid = threadIdx.x;
    float s = 0.f;
    for (int i = tid; i < NBLK; i += 256) s += ws[i];   // fixed order per thread
    sb[tid] = s;
    __syncthreads();
    if (tid == 0) {
        float tot = 0.f;
        for (int i = 0; i < 256; ++i) tot += sb[i];     // fixed order
        out[0] = tot / (float)NPIX;
    }
}

extern "C" void kernel_launch(void* const* d_in, const int* in_sizes, int n_in,
                              void* d_out, int out_size, void* d_ws, size_t ws_size,
                              hipStream_t stream) {
    (void)in_sizes; (void)n_in; (void)out_size; (void)ws_size;
    const float* img1 = (const float*)d_in[0];
    const float* img2 = (const float*)d_in[1];
    float* ws  = (float*)d_ws;
    float* out = (float*)d_out;

    ssim_tile_kernel<<<NBLK, 256, 0, stream>>>(img1, img2, ws);
    ssim_reduce_kernel<<<1, 256, 0, stream>>>(ws, out);
}